// EC_SA41_90082644067015
// MI455X (gfx1250) — compile-verified
//
#include <hip/hip_runtime.h>
#include <math.h>

// ---------------------------------------------------------------------------
// ECA-style channel+spatial attention, MI455X (gfx1250, wave32).
// Memory-bound: ~768 MB total traffic -> ~33 us at 23.3 TB/s.
//  - K1: one fused streaming pass over x (512 WGs for HBM latency coverage,
//        non-temporal loads), producing f1sum (b,c) and f2sum (b,p) partials.
//  - K3: spatial conv as 6x (64x64 @ 64x64) f32 WMMA matmuls per batch;
//        input tile staged into LDS by the Tensor Data Mover with the TDM
//        pad feature creating a 65-float row pitch (bank-conflict free).
//  - K4: streaming gate pass with non-temporal stores.
// ---------------------------------------------------------------------------

typedef float    v2f __attribute__((ext_vector_type(2)));
typedef float    v4f __attribute__((ext_vector_type(4)));
typedef float    v8f __attribute__((ext_vector_type(8)));
typedef unsigned v4u __attribute__((ext_vector_type(4)));
typedef int      v4i __attribute__((ext_vector_type(4)));
typedef int      v8i __attribute__((ext_vector_type(8)));

#define B_    32
#define C_    512
#define H_    64
#define W_    64
#define HW_   4096
#define NPIX  (B_ * HW_)          /* 131072 */
#define BN_EPS 1e-5f

__device__ __forceinline__ float sigmoidf_(float v) {
    return 1.0f / (1.0f + __expf(-v));
}

// ---------------------------------------------------------------------------
// K0: zero the atomically-accumulated regions: f1sum (16384) + stats (2) +
// pad + f2sum (131072) = [0, 147464) floats. d_ws is poisoned 0xAA.
// ---------------------------------------------------------------------------
__global__ void k0_zero(float* __restrict__ ws) {
    int i = blockIdx.x * blockDim.x + threadIdx.x;
    if (i < 147464) ws[i] = 0.0f;
}

// ---------------------------------------------------------------------------
// K1: fused reduction pass over x (256 MB, read exactly once, NT float4).
//   - f2sum[b,p] += sum over this block's 128-channel chunk (atomicAdd)
//   - f1sum[b,c]  = sum over p (per-c wave32 shfl reduction + atomicAdd)
// grid: 32 b * 4 ptiles * 4 c-chunks = 512 blocks of 256 threads
// (4096 waves -> enough outstanding 128B loads to cover HBM latency).
// ---------------------------------------------------------------------------
__global__ void __launch_bounds__(256) k1_reduce(const float* __restrict__ x,
                                                 float* __restrict__ f1sum,
                                                 float* __restrict__ f2sum) {
    const int b    = blockIdx.x >> 4;
    const int pt   = (blockIdx.x >> 2) & 3;
    const int cc   = blockIdx.x & 3;
    const int tid  = threadIdx.x;
    const int lane = tid & 31;
    const int p0   = pt * 1024 + tid * 4;          // pixel base for this thread
    const int c0   = cc * 128;
    const v4f* x4  = (const v4f*)x;

    v4f acc = {0.f, 0.f, 0.f, 0.f};
    #pragma unroll 4
    for (int c = c0; c < c0 + 128; ++c) {
        v4f v = __builtin_nontemporal_load(&x4[(b * C_ + c) * (HW_ / 4) + (p0 >> 2)]);
        acc += v;
        float s = (v[0] + v[1]) + (v[2] + v[3]);
        #pragma unroll
        for (int off = 16; off > 0; off >>= 1) s += __shfl_down(s, off, 32);
        if (lane == 0) atomicAdd(&f1sum[b * C_ + c], s);
    }
    float* dst = &f2sum[b * HW_ + p0];
    atomicAdd(&dst[0], acc[0]);
    atomicAdd(&dst[1], acc[1]);
    atomicAdd(&dst[2], acc[2]);
    atomicAdd(&dst[3], acc[3]);
}

// ---------------------------------------------------------------------------
// K2: channel attention: 5-tap conv over channel axis (pad 2) + sigmoid.
// ---------------------------------------------------------------------------
__global__ void k2_chattn(const float* __restrict__ f1sum,
                          const float* __restrict__ w1,
                          float* __restrict__ f1att) {
    int i = blockIdx.x * blockDim.x + threadIdx.x;   // i = b*512 + c
    if (i >= B_ * C_) return;
    const int b = i >> 9;
    const int c = i & 511;
    float r = 0.0f;
    #pragma unroll
    for (int k = 0; k < 5; ++k) {
        int cc = c + k - 2;
        if (cc >= 0 && cc < C_)
            r += w1[k] * f1sum[b * C_ + cc] * (1.0f / HW_);
    }
    f1att[i] = sigmoidf_(r);
}

// ---------------------------------------------------------------------------
// K3: spatial attention via WMMA f32 16x16x4.
//   f2conv = (1/512) * sum_k [ wh_k @ shift_col(S, k-1) + shift_row(S, k-1) @ ww_k^T ]
// where S = f2sum[b] (un-normalized channel sums; /512 folded into output).
// One workgroup per batch b; 8 waves, 2 output tiles (16x16) per wave.
// S staged into LDS by the Tensor Data Mover; D# pad feature (64 DWORD
// interval, 1 DWORD pad) yields a 65-float row pitch -> no bank conflicts
// on the column (shift_row) reads.
// Also accumulates global sum / sum^2 of f2conv for batch-norm.
// ---------------------------------------------------------------------------
__global__ void __launch_bounds__(256) k3_spatial(const float* __restrict__ f2sum,
                                                  const float* __restrict__ wh,
                                                  const float* __restrict__ ww,
                                                  float* __restrict__ f2conv,
                                                  float* __restrict__ stats) {
    __shared__ float Msh[64 * 65];                  // 65-float row pitch
    const int b   = blockIdx.x;
    const int tid = threadIdx.x;

#if defined(__has_builtin) && __has_builtin(__builtin_amdgcn_tensor_load_to_lds)
    if (tid < 32) {                                 // wave 0 only: EXEC ignored by
        // TDM ops, so keep the issue in a single wave's control flow.
        const unsigned long long ga = (unsigned long long)(const void*)(f2sum + b * HW_);
        const unsigned lds = (unsigned)(uintptr_t)(&Msh[0]);
        v4u g0;
        g0[0] = 1u;                                         // count=1, user mode
        g0[1] = lds;                                        // lds_addr
        g0[2] = (unsigned)(ga & 0xFFFFFFFFu);               // global_addr[31:0]
        g0[3] = (unsigned)((ga >> 32) & 0x01FFFFFFu)        // global_addr[56:32]
              | 0x80000000u;                                // type=2 (image)
        v8i g1;
        g1[0] = (int)((2u << 16)      // data_size = 4 bytes
                    | (1u << 20)      // pad_enable
                    | (5u << 22)      // pad_interval: 64 DWORDs (one 64-col row)
                    | (0u << 25));    // pad_amount: 1 DWORD -> 65-float pitch
        g1[1] = (int)(64u << 16);     // tensor_dim0 = 64
        g1[2] = (int)(64u << 16);     // tensor_dim1 = 64
        g1[3] = (int)(64u << 16);     // tile_dim0 = 64
        g1[4] = 64;                   // tile_dim1 = 64, tile_dim2 = 0
        g1[5] = 64;                   // tensor_dim0_stride = 64
        g1[6] = 0;                    // tensor_dim1_stride (unused, 2D)
        g1[7] = 0;
        v4i gz4 = {0, 0, 0, 0};
        v8i gz8 = {0, 0, 0, 0, 0, 0, 0, 0};
        // clang-23 / therock-10.0 6-arg form (probe-confirmed arity).
        __builtin_amdgcn_tensor_load_to_lds(g0, g1, gz4, gz4, gz8, 0);
        __builtin_amdgcn_s_wait_tensorcnt(0);
    }
#else
    for (int i = tid; i < 4096; i += 256) {
        Msh[(i >> 6) * 65 + (i & 63)] = f2sum[b * HW_ + i];
    }
#endif
    __syncthreads();

    const int wid  = tid >> 5;
    const int lane = tid & 31;
    const int mn   = lane & 15;                     // M (for A) or N (for B/C)
    const int half = lane >> 4;                     // selects K pair / row half

    float lsum = 0.0f, lss = 0.0f;

    for (int tt = 0; tt < 2; ++tt) {
        const int t  = wid + tt * 8;                // output tile 0..15
        const int ti = t >> 2;                      // row tile
        const int tj = t & 3;                       // col tile
        v8f acc = {0.f, 0.f, 0.f, 0.f, 0.f, 0.f, 0.f, 0.f};

        for (int kt = 0; kt < 3; ++kt) {            // conv tap
            const int s = kt - 1;                   // shift (cross-correlation)

            // term 1: wh_k[ho,hi] @ S[hi, w+s]   (column-shifted, zero pad)
            const int colB = tj * 16 + mn + s;
            const bool okB = (colB >= 0) && (colB < 64);
            #pragma unroll
            for (int kk = 0; kk < 16; ++kk) {
                const int k0 = kk * 4 + half * 2;   // K index of vgpr 0
                v2f a, bf;
                a[0] = wh[(ti * 16 + mn) * 192 + (k0 + 0) * 3 + kt];
                a[1] = wh[(ti * 16 + mn) * 192 + (k0 + 1) * 3 + kt];
                bf[0] = okB ? Msh[(k0 + 0) * 65 + colB] : 0.0f;
                bf[1] = okB ? Msh[(k0 + 1) * 65 + colB] : 0.0f;
                acc = __builtin_amdgcn_wmma_f32_16x16x4_f32(
                    false, a, false, bf, (short)0, acc, false, false);
            }

            // term 2: S[h+s, wi] @ ww_k^T[wi,wo]  (row-shifted, zero pad)
            const int rowA = ti * 16 + mn + s;
            const bool okA = (rowA >= 0) && (rowA < 64);
            #pragma unroll
            for (int kk = 0; kk < 16; ++kk) {
                const int k0 = kk * 4 + half * 2;
                v2f a, bf;
                a[0] = okA ? Msh[rowA * 65 + k0 + 0] : 0.0f;
                a[1] = okA ? Msh[rowA * 65 + k0 + 1] : 0.0f;
                bf[0] = ww[(tj * 16 + mn) * 192 + (k0 + 0) * 3 + kt];
                bf[1] = ww[(tj * 16 + mn) * 192 + (k0 + 1) * 3 + kt];
                acc = __builtin_amdgcn_wmma_f32_16x16x4_f32(
                    false, a, false, bf, (short)0, acc, false, false);
            }
        }

        #pragma unroll
        for (int r = 0; r < 8; ++r) {
            const int row = ti * 16 + r + half * 8;
            const int col = tj * 16 + mn;
            const float v = acc[r] * (1.0f / C_);   // fold channel-mean /512
            f2conv[b * HW_ + row * 64 + col] = v;
            lsum += v;
            lss  += v * v;
        }
    }

    #pragma unroll
    for (int off = 16; off > 0; off >>= 1) {
        lsum += __shfl_down(lsum, off, 32);
        lss  += __shfl_down(lss,  off, 32);
    }
    if (lane == 0) {
        atomicAdd(&stats[0], lsum);
        atomicAdd(&stats[1], lss);
    }
}

// ---------------------------------------------------------------------------
// K3b: batch-norm (training-mode batch stats) + sigmoid -> spatial gate sa.
// ---------------------------------------------------------------------------
__global__ void k3b_norm(const float* __restrict__ f2conv,
                         const float* __restrict__ stats,
                         const float* __restrict__ gamma,
                         const float* __restrict__ beta,
                         float* __restrict__ sa) {
    int i = blockIdx.x * blockDim.x + threadIdx.x;
    if (i >= NPIX) return;
    const float mu  = stats[0] * (1.0f / NPIX);
    const float var = stats[1] * (1.0f / NPIX) - mu * mu;
    const float v = gamma[0] * (f2conv[i] - mu) * rsqrtf(var + BN_EPS) + beta[0];
    sa[i] = sigmoidf_(v);
}

// ---------------------------------------------------------------------------
// K4: streaming output pass: out = x * f1att[b,c] * sa[b,p]  (float4).
// sa is 512 KB -> its 512x reuse comes from L2; x/out stream with NT stores.
// ---------------------------------------------------------------------------
__global__ void __launch_bounds__(256) k4_out(const float* __restrict__ x,
                                              const float* __restrict__ f1att,
                                              const float* __restrict__ sa,
                                              float* __restrict__ out) {
    const int i4    = blockIdx.x * blockDim.x + threadIdx.x;  // float4 index
    const int base  = i4 * 4;                                 // element index
    const int plane = base >> 12;                             // b*512 + c
    const int b     = plane >> 9;
    const int p     = base & 4095;

    const v4f xv = __builtin_nontemporal_load(&((const v4f*)x)[i4]);
    const float g = f1att[plane];
    const v4f sv = ((const v4f*)sa)[(b << 10) + (p >> 2)];

    v4f o = xv * g * sv;
    __builtin_nontemporal_store(o, &((v4f*)out)[i4]);
}

// ---------------------------------------------------------------------------
// Launch
// ---------------------------------------------------------------------------
extern "C" void kernel_launch(void* const* d_in, const int* in_sizes, int n_in,
                              void* d_out, int out_size, void* d_ws, size_t ws_size,
                              hipStream_t stream) {
    const float* x     = (const float*)d_in[0];
    const float* w1    = (const float*)d_in[1];
    const float* wh    = (const float*)d_in[2];
    const float* ww    = (const float*)d_in[3];
    const float* gamma = (const float*)d_in[4];
    const float* beta  = (const float*)d_in[5];

    float* ws     = (float*)d_ws;
    float* f1sum  = ws;               // 16384 floats: (b,c) h/w-sums
    float* stats  = ws + 16384;       // 2 floats: sum, sum^2 of f2conv
    float* f2sum  = ws + 16392;       // 131072 floats: (b,p) channel sums (16B aligned)
    float* f1att  = ws + 147464;      // 16384 floats: channel gate
    float* f2conv = ws + 163848;      // 131072 floats
    float* sa     = ws + 294920;      // 131072 floats: spatial gate
    float* out    = (float*)d_out;

    hipLaunchKernelGGL(k0_zero,    dim3(577),   dim3(256), 0, stream, ws);
    hipLaunchKernelGGL(k1_reduce,  dim3(512),   dim3(256), 0, stream, x, f1sum, f2sum);
    hipLaunchKernelGGL(k2_chattn,  dim3(64),    dim3(256), 0, stream, f1sum, w1, f1att);
    hipLaunchKernelGGL(k3_spatial, dim3(32),    dim3(256), 0, stream, f2sum, wh, ww, f2conv, stats);
    hipLaunchKernelGGL(k3b_norm,   dim3(512),   dim3(256), 0, stream, f2conv, stats, gamma, beta, sa);
    hipLaunchKernelGGL(k4_out,     dim3(65536), dim3(256), 0, stream, x, f1att, sa, out);
}